// SumGoalHistory_19387482374813
// MI455X (gfx1250) — compile-verified
//
#include <hip/hip_runtime.h>
#include <hip/hip_bf16.h>

// T=512, B=64, D=1024 ; BD = 65536 columns ; 8 chunks x 64 timesteps.
// Phase A: per-(chunk, col4) partial sums            (131072 threads)
// Phase B: exclusive scan of chunk sums + init, writes final_state   (16384 threads)
// Phase C: WMMA triangular-matrix scan within chunks + sigmoid (32768 waves)

typedef float v8f __attribute__((ext_vector_type(8)));
typedef float v2f __attribute__((ext_vector_type(2)));

#define SGH_T   512
#define SGH_BD  65536
#define SGH_CH  8      // chunks
#define SGH_CL  64     // chunk length (timesteps)

// ---------------- Phase A: chunk partial sums (float4 per thread) -------------
__global__ __launch_bounds__(256) void sgh_chunk_sums(const float* __restrict__ goal,
                                                      float* __restrict__ partials) {
    const int Q = SGH_BD / 4;                       // 16384 float4 columns
    int idx   = blockIdx.x * 256 + threadIdx.x;     // 0 .. 8*16384-1
    int col4  = idx & (Q - 1);
    int chunk = idx >> 14;
    const float4* g = reinterpret_cast<const float4*>(goal) + (size_t)chunk * SGH_CL * Q + col4;
    float4 acc = make_float4(0.f, 0.f, 0.f, 0.f);
#pragma unroll 16
    for (int t = 0; t < SGH_CL; ++t) {
        float4 v = g[(size_t)t * Q];
        acc.x += v.x; acc.y += v.y; acc.z += v.z; acc.w += v.w;
    }
    reinterpret_cast<float4*>(partials)[(size_t)chunk * Q + col4] = acc;
}

// -------- Phase B: in-place exclusive scan of chunk sums (+init); final state --
__global__ __launch_bounds__(256) void sgh_scan_chunks(const float* __restrict__ init,
                                                       float* __restrict__ partials,
                                                       float* __restrict__ out) {
    const int Q = SGH_BD / 4;
    int col4 = blockIdx.x * 256 + threadIdx.x;      // 0 .. 16383
    float4*       P = reinterpret_cast<float4*>(partials);
    const float4* I = reinterpret_cast<const float4*>(init);
    float4 run = I[col4];
#pragma unroll
    for (int c = 0; c < SGH_CH; ++c) {
        float4 p = P[(size_t)c * Q + col4];
        P[(size_t)c * Q + col4] = run;              // exclusive offset incl. init
        run.x += p.x; run.y += p.y; run.z += p.z; run.w += p.w;
    }
    // final_state = init + total sum, written twice (tuple returns it twice)
    float4* O = reinterpret_cast<float4*>(out);
    O[(size_t)SGH_T * Q + col4]     = run;
    O[(size_t)SGH_T * Q + Q + col4] = run;
}

// ---- Phase C: within-chunk scan via V_WMMA_F32_16X16X4_F32 (D = L16*G + C) ----
// One wave32 owns a 16-column x 64-timestep tile. 4 sequential 16x16 tiles,
// carry passed via ds_bpermute of C row M=15 (VGPR 7, lanes 16-31).
__global__ __launch_bounds__(256) void sgh_scan_wmma(const float* __restrict__ goal,
                                                     const float* __restrict__ offs,
                                                     float* __restrict__ out) {
    int tid  = threadIdx.x;
    int lane = tid & 31;
    int w       = blockIdx.x * 8 + (tid >> 5);      // 0 .. 32767
    int coltile = w & 4095;
    int chunk   = w >> 12;
    int col0    = coltile << 4;
    int n       = lane & 15;                        // column within tile (N)
    int hi      = (lane >> 4) & 1;                  // lane-half
    int rsel    = hi << 1;                          // K-offset of this lane-half {0,2}
    int m       = n;                                // A-matrix row (M = lane%16)

    // A = lower-triangular ones, split into four 16x4 K-slices.
    // A layout (32-bit 16x4): lanes 0-15 hold K={0,1} in VGPR{0,1}; lanes 16-31 K={2,3}.
    // Pin the values behind an asm barrier so they're built once, not
    // rematerialized (cmp+cndmask) inside every tile iteration.
    float ax[4], ay[4];
#pragma unroll
    for (int k = 0; k < 4; ++k) {
        int kk = 4 * k + rsel;
        ax[k] = (kk     <= m) ? 1.0f : 0.0f;
        ay[k] = (kk + 1 <= m) ? 1.0f : 0.0f;
    }
    asm("" : "+v"(ax[0]), "+v"(ax[1]), "+v"(ax[2]), "+v"(ax[3]),
             "+v"(ay[0]), "+v"(ay[1]), "+v"(ay[2]), "+v"(ay[3]));
    v2f a[4];
#pragma unroll
    for (int k = 0; k < 4; ++k) { a[k].x = ax[k]; a[k].y = ay[k]; }

    float carry = offs[(size_t)chunk * SGH_BD + col0 + n];
    int tbase = chunk * SGH_CL;

    // ---- stage all B-matrix data for this 64-timestep chunk (32 loads) ----
    // B 4x16 layout mirrors A: lanes 0-15 rows K={4k,4k+1}, lanes 16-31 K={4k+2,4k+3}
    // Last use of goal: non-temporal loads (still hit L2 lines from phase A,
    // but flag them for early eviction so the output stream has room).
    const float* bbase = goal + (size_t)(tbase + rsel) * SGH_BD + col0 + n;
    float bx[4][4], by[4][4];                       // [tile][k]
#pragma unroll
    for (int tile = 0; tile < 4; ++tile) {
        const float* pt = bbase + (size_t)tile * (16 * SGH_BD);
#pragma unroll
        for (int k = 0; k < 4; ++k) {
            bx[tile][k] = __builtin_nontemporal_load(pt + (size_t)(4 * k)     * SGH_BD);
            by[tile][k] = __builtin_nontemporal_load(pt + (size_t)(4 * k + 1) * SGH_BD);
        }
    }

    float* obase = out + ((size_t)tbase + (size_t)hi * 8) * SGH_BD + col0 + n;

#pragma unroll
    for (int tile = 0; tile < 4; ++tile) {
        v8f c;
#pragma unroll
        for (int v = 0; v < 8; ++v) c[v] = carry;   // C(m,n) = carry(n) for all m

#pragma unroll
        for (int k = 0; k < 4; ++k) {
            v2f b;
            b.x = bx[tile][k];
            b.y = by[tile][k];
            c = __builtin_amdgcn_wmma_f32_16x16x4_f32(
                    /*neg_a=*/false, a[k], /*neg_b=*/false, b,
                    /*c_mod=*/(short)0, c, /*reuse_a=*/false, /*reuse_b=*/false);
        }

        // new carry(n) = D(15,n): C-VGPR 7, lanes 16-31 (M=15). Broadcast by column.
        carry = __builtin_bit_cast(float,
                    __builtin_amdgcn_ds_bpermute((16 + n) << 2,
                                                 __builtin_bit_cast(int, c[7])));

        // sigmoid via v_exp_f32 + v_rcp_f32 (avoid the exact-division Newton
        // chain the compiler emits for 1.0f/x) + non-temporal stores.
        float* ot = obase + (size_t)tile * (16 * SGH_BD);
#pragma unroll
        for (int v = 0; v < 8; ++v) {               // C layout: M = v (+8 for hi lanes)
            float e = __expf(-c[v]);
            float s = __builtin_amdgcn_rcpf(1.0f + e);
            __builtin_nontemporal_store(s, ot + (size_t)v * SGH_BD);
        }
    }
}

extern "C" void kernel_launch(void* const* d_in, const int* in_sizes, int n_in,
                              void* d_out, int out_size, void* d_ws, size_t ws_size,
                              hipStream_t stream) {
    const float* goal = (const float*)d_in[0];     // [512, 64, 1024]
    const float* init = (const float*)d_in[1];     // [1, 64, 1024]
    float* out = (float*)d_out;                    // outputs | final | final
    float* partials = (float*)d_ws;                // 8 * 65536 floats = 2 MB

    (void)in_sizes; (void)n_in; (void)out_size; (void)ws_size;

    // Phase A: 8*16384 threads
    sgh_chunk_sums<<<512, 256, 0, stream>>>(goal, partials);
    // Phase B: 16384 threads (also writes final_state twice)
    sgh_scan_chunks<<<64, 256, 0, stream>>>(init, partials, out);
    // Phase C: 4096 blocks x 8 waves = 32768 wave-tiles
    sgh_scan_wmma<<<4096, 256, 0, stream>>>(goal, partials, out);
}